// CapsLayer_41154376630999
// MI455X (gfx1250) — compile-verified
//
#include <hip/hip_runtime.h>
#include <math.h>

typedef float v2f __attribute__((ext_vector_type(2)));
typedef float v4f __attribute__((ext_vector_type(4)));
typedef float v8f __attribute__((ext_vector_type(8)));

#define N_BATCH 512
#define N_IN    1152
#define N_OUT   10
#define O_DIM   16
#define I_DIM   8
#define BT      16                    // batch tile (WMMA M)
#define NWAVES  16
#define THREADS (NWAVES * 32)
#define I_PER_WAVE (N_IN / NWAVES)    // 72
#define TILE_F  (N_OUT * 256)         // 2560 floats: 10 o-tiles of 16x16

// ---- LDS layout (float offsets), 56320 floats = 220 KB < 320 KB/WGP ----
#define STAGE_OFF 0                              // per-wave u_hat staging / s partials
#define STAGE_SZ  (NWAVES * TILE_F)              // 40960
#define V_OFF     (STAGE_OFF + STAGE_SZ)         // v for r=0,1,2: [(b*10+o)*16+j]
#define V_SZ      (3 * TILE_F)                   // 7680
#define S_OFF     (V_OFF + V_SZ)                 // reduced s
#define S_SZ      TILE_F                         // 2560
#define ABUF_OFF  (S_OFF + S_SZ)                 // per-wave agreement exchange [b][o]
#define ABUF_SZ   (NWAVES * BT * N_OUT)          // 2560
#define CBUF_OFF  (ABUF_OFF + ABUF_SZ)           // per-wave coupling coeffs [o][b]
#define CBUF_SZ   (NWAVES * N_OUT * BT)          // 2560
#define LDS_FLOATS (CBUF_OFF + CBUF_SZ)          // 56320

extern "C" __global__ __launch_bounds__(THREADS)
void caps_routing_kernel(const float* __restrict__ u,
                         const float* __restrict__ W,
                         float* __restrict__ out)
{
    extern __shared__ float smem[];
    const int tid  = threadIdx.x;
    const int wave = tid >> 5;
    const int lane = tid & 31;
    const int half = lane >> 4;     // lane half selects K-pair (A) / K-row (B)
    const int m    = lane & 15;     // A: batch row; B: out_dim column

    const int tile = blockIdx.x;    // 16-batch tile
    const float* uA = u + (size_t)(tile * BT + m) * N_IN * I_DIM;

    float* stage = smem + STAGE_OFF + wave * TILE_F;
    float* vbuf  = smem + V_OFF;
    float* sbuf  = smem + S_OFF;
    float* abuf  = smem + ABUF_OFF + wave * (BT * N_OUT);
    float* cbuf  = smem + CBUF_OFF + wave * (N_OUT * BT);

    const int i0 = wave * I_PER_WAVE;
    const v8f vzero = {0.f, 0.f, 0.f, 0.f, 0.f, 0.f, 0.f, 0.f};

    for (int r = 0; r < 3; ++r) {
        v8f sacc[N_OUT];
        #pragma unroll
        for (int o = 0; o < N_OUT; ++o) sacc[o] = vzero;

        for (int ii = 0; ii < I_PER_WAVE; ++ii) {
            const int i = i0 + ii;
            // A fragments (16x4 f32, two K-halves): lanes 0-15 d={0,1}/{4,5}, 16-31 d={2,3}/{6,7}
            v2f a_lo = *(const v2f*)(uA + (size_t)i * I_DIM + 2 * half);
            v2f a_hi = *(const v2f*)(uA + (size_t)i * I_DIM + 4 + 2 * half);
            const float* Wi = W + (size_t)i * N_OUT * O_DIM * I_DIM;
            if (ii + 1 < I_PER_WAVE)
                __builtin_prefetch(Wi + N_OUT * O_DIM * I_DIM, 0, 3); // next capsule's weights

            if (r == 0) {
                // uniform coupling: s = 0.1 * sum_i u_hat -> chain WMMA accumulate
                #pragma unroll
                for (int o = 0; o < N_OUT; ++o) {
                    const float* Wo = Wi + o * O_DIM * I_DIM;
                    v2f b_lo = *(const v2f*)(Wo + m * I_DIM + 2 * half);
                    v2f b_hi = *(const v2f*)(Wo + m * I_DIM + 4 + 2 * half);
                    sacc[o] = __builtin_amdgcn_wmma_f32_16x16x4_f32(
                        false, a_lo, false, b_lo, (short)0, sacc[o], false, false);
                    sacc[o] = __builtin_amdgcn_wmma_f32_16x16x4_f32(
                        false, a_hi, false, b_hi, (short)0, sacc[o], false, false);
                }
            } else {
                // u_hat tiles for this capsule i, staged to per-wave LDS
                #pragma unroll
                for (int o = 0; o < N_OUT; ++o) {
                    const float* Wo = Wi + o * O_DIM * I_DIM;
                    v2f b_lo = *(const v2f*)(Wo + m * I_DIM + 2 * half);
                    v2f b_hi = *(const v2f*)(Wo + m * I_DIM + 4 + 2 * half);
                    v8f uh = __builtin_amdgcn_wmma_f32_16x16x4_f32(
                        false, a_lo, false, b_lo, (short)0, vzero, false, false);
                    uh = __builtin_amdgcn_wmma_f32_16x16x4_f32(
                        false, a_hi, false, b_hi, (short)0, uh, false, false);
                    v4f lo = {uh[0], uh[1], uh[2], uh[3]};
                    v4f hi = {uh[4], uh[5], uh[6], uh[7]};
                    *(v4f*)(stage + o * 256 + lane * 8)     = lo;
                    *(v4f*)(stage + o * 256 + lane * 8 + 4) = hi;
                }
                asm volatile("s_wait_dscnt 0x0" ::: "memory"); // wave-internal LDS visibility

                // agreements b[i][o] = sum_{rr<r} <u_hat, v_rr>, lane -> (b=m, 5 o's per half)
                const int kk = m & 7, hh = m >> 3;
                #pragma unroll
                for (int oo = 0; oo < 5; ++oo) {
                    const int o = half * 5 + oo;
                    const float* uhp = stage + o * 256 + hh * 128 + kk; // element (m, j) at +j*8
                    const float* v0p = vbuf + (m * N_OUT + o) * O_DIM;
                    const float* v1p = v0p + TILE_F;
                    float acc = 0.f;
                    #pragma unroll
                    for (int j = 0; j < O_DIM; ++j) {
                        const float x = uhp[j * 8];
                        acc += x * v0p[j];
                        if (r == 2) acc += x * v1p[j];
                    }
                    abuf[m * N_OUT + o] = acc;
                }
                asm volatile("s_wait_dscnt 0x0" ::: "memory");

                // softmax over n_out (both halves compute full row b=m)
                float bv[N_OUT], mx = -3.402823e38f;
                #pragma unroll
                for (int o = 0; o < N_OUT; ++o) {
                    bv[o] = abuf[m * N_OUT + o];
                    mx = fmaxf(mx, bv[o]);
                }
                float sum = 0.f;
                #pragma unroll
                for (int o = 0; o < N_OUT; ++o) { bv[o] = expf(bv[o] - mx); sum += bv[o]; }
                const float inv = 1.f / sum;
                #pragma unroll
                for (int oo = 0; oo < 5; ++oo) {
                    const int o = half * 5 + oo;
                    cbuf[o * BT + m] = bv[o] * inv; // [o][b] for broadcast-friendly reads
                }
                asm volatile("s_wait_dscnt 0x0" ::: "memory");

                // s += c * u_hat in D-register layout (c uniform per lane-half rows)
                #pragma unroll
                for (int o = 0; o < N_OUT; ++o) {
                    v4f c0 = *(const v4f*)(cbuf + o * BT + half * 8);
                    v4f c1 = *(const v4f*)(cbuf + o * BT + half * 8 + 4);
                    v4f u0 = *(const v4f*)(stage + o * 256 + lane * 8);
                    v4f u1 = *(const v4f*)(stage + o * 256 + lane * 8 + 4);
                    #pragma unroll
                    for (int k = 0; k < 4; ++k) {
                        sacc[o][k]     += c0[k] * u0[k];
                        sacc[o][k + 4] += c1[k] * u1[k];
                    }
                }
            }
        }

        // dump per-wave s partials into own staging slab (wave-private, DS in-order)
        #pragma unroll
        for (int o = 0; o < N_OUT; ++o) {
            v4f lo = {sacc[o][0], sacc[o][1], sacc[o][2], sacc[o][3]};
            v4f hi = {sacc[o][4], sacc[o][5], sacc[o][6], sacc[o][7]};
            *(v4f*)(stage + o * 256 + lane * 8)     = lo;
            *(v4f*)(stage + o * 256 + lane * 8 + 4) = hi;
        }
        __syncthreads();

        // cross-wave reduce -> s[b][o][j]
        const float cscale = (r == 0) ? 0.1f : 1.0f;
        for (int t = tid; t < TILE_F; t += THREADS) {
            const int o  = t / 256;
            const int ln = (t & 255) >> 3;
            const int k  = t & 7;
            const int b  = k + 8 * (ln >> 4);
            const int j  = ln & 15;
            float s = 0.f;
            #pragma unroll
            for (int w = 0; w < NWAVES; ++w)
                s += smem[STAGE_OFF + w * TILE_F + t];
            sbuf[(b * N_OUT + o) * O_DIM + j] = s * cscale;
        }
        __syncthreads();

        // squash -> v_r
        if (tid < BT * N_OUT) {
            const int b = tid / N_OUT, o = tid % N_OUT;
            const float* sp = sbuf + (b * N_OUT + o) * O_DIM;
            float sq = 0.f;
            #pragma unroll
            for (int j = 0; j < O_DIM; ++j) sq += sp[j] * sp[j];
            const float sc = sq / ((1.f + sq) * sqrtf(sq + 1e-6f));
            float* vp = vbuf + r * TILE_F + (b * N_OUT + o) * O_DIM;
            #pragma unroll
            for (int j = 0; j < O_DIM; ++j) vp[j] = sp[j] * sc;
        }
        __syncthreads();
    }

    // final v (iteration 2) -> out[b][o][j], already contiguous per tile
    float* outp = out + (size_t)tile * TILE_F;
    for (int t = tid; t < TILE_F; t += THREADS)
        outp[t] = vbuf[2 * TILE_F + t];
}

extern "C" void kernel_launch(void* const* d_in, const int* in_sizes, int n_in,
                              void* d_out, int out_size, void* d_ws, size_t ws_size,
                              hipStream_t stream) {
    (void)in_sizes; (void)n_in; (void)out_size; (void)d_ws; (void)ws_size;
    const float* u = (const float*)d_in[0];   // [512,1152,8] f32
    const float* W = (const float*)d_in[1];   // [1152,10,16,8] f32
    float* out = (float*)d_out;               // [512,10,16] f32

    static_assert(LDS_FLOATS * sizeof(float) <= 320 * 1024, "exceeds WGP LDS");
    const size_t shmem = LDS_FLOATS * sizeof(float);
    // opt in to >64KB dynamic LDS (gfx1250 WGP has 320KB); ignore failure
    (void)hipFuncSetAttribute((const void*)caps_routing_kernel,
                              hipFuncAttributeMaxDynamicSharedMemorySize, (int)shmem);
    caps_routing_kernel<<<dim3(N_BATCH / BT), dim3(THREADS), shmem, stream>>>(u, W, out);
}